// KnowledgeIntegrator_33011118637185
// MI455X (gfx1250) — compile-verified
//
#include <hip/hip_runtime.h>
#include <math.h>

#define B_   32
#define S_   128
#define D_   768
#define C_   25
#define KB_  100
#define TOPK_ 10
#define NEGV (-1e9f)
#define TOK_ (B_ * S_)   /* 4096 tokens */
#define K2D_ (2 * D_)    /* 1536 */
#define NT_  7           /* ceil(100/16) N tiles */
#define KC_  128         /* K-chunk rows staged in LDS */

typedef __attribute__((ext_vector_type(2))) float v2f;
typedef __attribute__((ext_vector_type(8))) float v8f;

// ---------------------------------------------------------------------------
// Kernel A: q = [slot|intent] @ W   (M=4096, K=1536, N=100), fp32 WMMA 16x16x4
// grid = (32 Mblocks, 1, 2 {W_know,W_ctx}), block = 256 (8 waves).
// Each wave owns one 16-row M tile and ALL 7 N tiles (7 x v8f accumulators),
// so each A fragment is loaded once and reused across 7 WMMAs.
// The block stages the full 112-col W panel in LDS, 128 K-rows at a time
// (128*112*4 = 56 KB). 768/128 = 6 => slot/intent boundary is chunk-aligned.
// ---------------------------------------------------------------------------
__global__ __launch_bounds__(256) void gemm_qw(
    const float* __restrict__ slot, const float* __restrict__ intent,
    const float* __restrict__ Wk,   const float* __restrict__ Wc,
    float* __restrict__ qout,       float* __restrict__ qcout)
{
    __shared__ float ldsB[NT_ * KC_ * 16];   // [ntile][krow][16] = 56 KB

    const int tid  = threadIdx.x;
    const int lane = tid & 31;
    const int wv   = tid >> 5;
    const int hi   = lane >> 4;   // 0: K pair {0,1}; 1: K pair {2,3}
    const int lo   = lane & 15;
    const int m0   = (blockIdx.x * 8 + wv) * 16;

    const float* W   = (blockIdx.z == 0) ? Wk   : Wc;
    float*       out = (blockIdx.z == 0) ? qout : qcout;

    v8f acc[NT_];
    #pragma unroll
    for (int nt = 0; nt < NT_; ++nt) acc[nt] = (v8f){};

    const int mrow = m0 + lo;

    for (int kc = 0; kc < K2D_ / KC_; ++kc) {
        // stage W[kc*128 .. +127][0..111] into LDS as [nt][kr][16], zero-pad
        for (int i = tid; i < NT_ * KC_ * 16; i += 256) {
            const int nt   = i >> 11;          // /2048
            const int rem  = i & 2047;
            const int kr   = rem >> 4;
            const int nn16 = rem & 15;
            const int col  = nt * 16 + nn16;
            ldsB[i] = (col < KB_) ? W[(kc * KC_ + kr) * KB_ + col] : 0.0f;
        }
        __syncthreads();

        // A base is uniform per chunk: chunks 0-5 -> slot, 6-11 -> intent
        const float* Abase = (kc < 6)
            ? (slot   + (long)mrow * D_ + kc * KC_)
            : (intent + (long)mrow * D_ + (kc - 6) * KC_);

        for (int kk = 0; kk < KC_ / 4; ++kk) {
            const int kl = kk * 4 + hi * 2;              // K pair within chunk
            v2f a = *reinterpret_cast<const v2f*>(Abase + kl);

            #pragma unroll
            for (int nt = 0; nt < NT_; ++nt) {
                v2f b;
                b.x = ldsB[nt * (KC_ * 16) + (kl + 0) * 16 + lo];
                b.y = ldsB[nt * (KC_ * 16) + (kl + 1) * 16 + lo];
                acc[nt] = __builtin_amdgcn_wmma_f32_16x16x4_f32(
                    false, a, false, b, (short)0, acc[nt], false, false);
            }
        }
        __syncthreads();
    }

    #pragma unroll
    for (int nt = 0; nt < NT_; ++nt) {
        const int col = nt * 16 + lo;
        if (col < KB_) {
            #pragma unroll
            for (int i = 0; i < 8; ++i) {
                const int row = m0 + i + hi * 8;
                out[(long)row * KB_ + col] = acc[nt][i];
            }
        }
    }
}

// ---------------------------------------------------------------------------
// Kernel B: per-token knowledge attention. One wave (32 threads) per token.
// scores -> exact top-10 threshold -> softmax -> weighted sum + pos embed.
// ---------------------------------------------------------------------------
__global__ __launch_bounds__(32) void know_attn(
    const float* __restrict__ q,
    const int* __restrict__ wn_idx,   const int* __restrict__ wn_len,
    const int* __restrict__ nell_idx, const int* __restrict__ nell_len,
    const float* __restrict__ wn_tab, const float* __restrict__ nell_tab,
    const float* __restrict__ pos,    float* __restrict__ knowledge)
{
    __shared__ float sc[50];
    __shared__ float pr[50];
    __shared__ int   cidx[50];

    const int tok  = blockIdx.x;
    const int s    = tok & (S_ - 1);
    const int lane = threadIdx.x;

    float qk[4];
    #pragma unroll
    for (int j = 0; j < 4; ++j) {
        const int k = lane + 32 * j;
        qk[j] = (k < KB_) ? q[(long)tok * KB_ + k] : 0.0f;
    }

    // scores over 50 concepts (wave-cooperative dot of length 100)
    for (int c = 0; c < 50; ++c) {
        const int idx = (c < C_) ? wn_idx[tok * C_ + c]
                                 : nell_idx[tok * C_ + (c - C_)];
        const float* row = ((c < C_) ? wn_tab : nell_tab) + (long)idx * KB_;
        float partial = 0.0f;
        #pragma unroll
        for (int j = 0; j < 4; ++j) {
            const int k = lane + 32 * j;
            if (k < KB_) partial += qk[j] * row[k];
        }
        for (int off = 16; off > 0; off >>= 1)
            partial += __shfl_xor(partial, off, 32);
        if (lane == 0) {
            sc[c]   = (idx != 0) ? partial : NEGV;
            cidx[c] = idx;
        }
    }
    __syncthreads();

    // exact 10th-largest threshold: T = min{ s_i : #{j : s_j > s_i} < 10 }
    float cand = INFINITY;
    #pragma unroll
    for (int t = 0; t < 2; ++t) {
        const int i = lane + 32 * t;
        if (i < 50) {
            const float si = sc[i];
            int cgt = 0;
            for (int j = 0; j < 50; ++j) cgt += (sc[j] > si) ? 1 : 0;
            if (cgt < TOPK_) cand = fminf(cand, si);
        }
    }
    for (int off = 16; off > 0; off >>= 1)
        cand = fminf(cand, __shfl_xor(cand, off, 32));
    const float T = cand;

    // masked softmax over 50
    float mx = NEGV;
    #pragma unroll
    for (int t = 0; t < 2; ++t) {
        const int i = lane + 32 * t;
        if (i < 50) {
            const float m = (sc[i] >= T) ? sc[i] : NEGV;
            mx = fmaxf(mx, m);
        }
    }
    for (int off = 16; off > 0; off >>= 1)
        mx = fmaxf(mx, __shfl_xor(mx, off, 32));

    float sum = 0.0f;
    #pragma unroll
    for (int t = 0; t < 2; ++t) {
        const int i = lane + 32 * t;
        if (i < 50) {
            const float m = (sc[i] >= T) ? sc[i] : NEGV;
            const float e = expf(m - mx);
            pr[i] = e;
            sum += e;
        }
    }
    for (int off = 16; off > 0; off >>= 1)
        sum += __shfl_xor(sum, off, 32);
    __syncthreads();
    const float inv = 1.0f / sum;

    // knowledge = probs @ concept_vecs (index-0 rows are zero => exact)
    float accv[4] = {0.0f, 0.0f, 0.0f, 0.0f};
    for (int c = 0; c < 50; ++c) {
        const float p = pr[c] * inv;
        const float* row = ((c < C_) ? wn_tab : nell_tab) + (long)cidx[c] * KB_;
        #pragma unroll
        for (int j = 0; j < 4; ++j) {
            const int k = lane + 32 * j;
            if (k < KB_) accv[j] += p * row[k];
        }
    }

    const int klen = wn_len[tok] + nell_len[tok];
    #pragma unroll
    for (int j = 0; j < 4; ++j) {
        const int k = lane + 32 * j;
        if (k < KB_) {
            float v = accv[j];
            if (klen != 0) v += pos[s * KB_ + k];
            knowledge[(long)tok * KB_ + k] = v;
        }
    }
}

// ---------------------------------------------------------------------------
// Kernel C: context attention + LayerNorm. One 128-thread block per (b,q).
// ---------------------------------------------------------------------------
__global__ __launch_bounds__(128) void ctx_attn(
    const float* __restrict__ qc,     const float* __restrict__ know,
    const int* __restrict__ wn_len,   const int* __restrict__ nell_len,
    const float* __restrict__ gamma,  const float* __restrict__ beta,
    float* __restrict__ ctx_out)
{
    __shared__ float qcl[KB_];
    __shared__ float sc[S_];
    __shared__ float red[S_];
    __shared__ float ctxv[S_];

    const int tok = blockIdx.x;
    const int b   = tok >> 7;
    const int t   = threadIdx.x;

    if (t < KB_) qcl[t] = qc[(long)tok * KB_ + t];
    __syncthreads();

    {   // scores over target positions
        const int klen = wn_len[b * S_ + t] + nell_len[b * S_ + t];
        const float* krow = know + (long)(b * S_ + t) * KB_;
        float dot = 0.0f;
        for (int k = 0; k < KB_; ++k) dot += qcl[k] * krow[k];
        sc[t] = (klen != 0) ? dot : NEGV;
    }
    __syncthreads();

    // softmax over 128
    red[t] = sc[t];
    __syncthreads();
    for (int off = 64; off > 0; off >>= 1) {
        if (t < off) red[t] = fmaxf(red[t], red[t + off]);
        __syncthreads();
    }
    const float mx = red[0];
    __syncthreads();
    const float e = expf(sc[t] - mx);
    sc[t]  = e;
    red[t] = e;
    __syncthreads();
    for (int off = 64; off > 0; off >>= 1) {
        if (t < off) red[t] += red[t + off];
        __syncthreads();
    }
    const float inv = 1.0f / red[0];
    __syncthreads();

    // contexts: thread t handles feature k=t (k<100)
    float cval = 0.0f;
    if (t < KB_) {
        for (int tt = 0; tt < S_; ++tt)
            cval += sc[tt] * know[(long)(b * S_ + tt) * KB_ + t];
        cval *= inv;
    }
    ctxv[t] = (t < KB_) ? cval : 0.0f;
    __syncthreads();

    // LayerNorm over 100 features
    red[t] = ctxv[t];
    __syncthreads();
    for (int off = 64; off > 0; off >>= 1) {
        if (t < off) red[t] += red[t + off];
        __syncthreads();
    }
    const float mu = red[0] / (float)KB_;
    __syncthreads();
    const float d = (t < KB_) ? (ctxv[t] - mu) : 0.0f;
    red[t] = d * d;
    __syncthreads();
    for (int off = 64; off > 0; off >>= 1) {
        if (t < off) red[t] += red[t + off];
        __syncthreads();
    }
    const float var = red[0] / (float)KB_;
    if (t < KB_) {
        ctx_out[(long)tok * KB_ + t] =
            (ctxv[t] - mu) * rsqrtf(var + 1e-5f) * gamma[t] + beta[t];
    }
}

// ---------------------------------------------------------------------------
extern "C" void kernel_launch(void* const* d_in, const int* in_sizes, int n_in,
                              void* d_out, int out_size, void* d_ws, size_t ws_size,
                              hipStream_t stream) {
    const float* intent   = (const float*)d_in[0];
    const float* slot     = (const float*)d_in[1];
    /* d_in[2] attention_mask: unused by the reference */
    const int*   wn_idx   = (const int*)d_in[3];
    const int*   wn_len   = (const int*)d_in[4];
    const int*   nell_idx = (const int*)d_in[5];
    const int*   nell_len = (const int*)d_in[6];
    const float* wn_tab   = (const float*)d_in[7];
    const float* nell_tab = (const float*)d_in[8];
    const float* Wk       = (const float*)d_in[9];
    const float* Wc       = (const float*)d_in[10];
    const float* gamma    = (const float*)d_in[11];
    const float* beta     = (const float*)d_in[12];
    const float* pos      = (const float*)d_in[13];

    float* out       = (float*)d_out;
    float* knowledge = out;                    // [4096*100]
    float* contexts  = out + (long)TOK_ * KB_; // [4096*100]

    float* q  = (float*)d_ws;                  // [4096*100]
    float* qc = q + (long)TOK_ * KB_;          // [4096*100]

    dim3 gA(TOK_ / (16 * 8), 1, 2);            // (32, 1, 2)
    gemm_qw<<<gA, 256, 0, stream>>>(slot, intent, Wk, Wc, q, qc);
    know_attn<<<TOK_, 32, 0, stream>>>(q, wn_idx, wn_len, nell_idx, nell_len,
                                       wn_tab, nell_tab, pos, knowledge);
    ctx_attn<<<TOK_, 128, 0, stream>>>(qc, knowledge, wn_len, nell_len,
                                       gamma, beta, contexts);
}